// TimeFold_33311766347849
// MI455X (gfx1250) — compile-verified
//
#include <hip/hip_runtime.h>

// ---------------------------------------------------------------------------
// TimeFold: irfft(1022) -> OLA -> segment crossfade fold -> rfft(1022)
// Both DFTs as dense f32 GEMMs on V_WMMA_F32_16X16X4_F32, 2x2 wave tiling.
// DFT matrices are generated on-device TRANSPOSED ([n][k]) so every A and B
// fragment is one contiguous ds_load_b64 (no register shuffles).
// ---------------------------------------------------------------------------

typedef float v2f __attribute__((ext_vector_type(2)));
typedef float v8f __attribute__((ext_vector_type(8)));

#define B_N        256
#define F_N        512
#define T_N        81
#define WIN_N      1022
#define HOP_N      400
#define NF_N       69
#define KEPT_LEN   30000
#define OUT_LEN    28550
#define SIG_STRIDE 33024          // >= 33022, multiple of 32
#define OUT_STRIDE 28560          // >= 28550, multiple of 16
#define KPAD       1024           // padded K (and N) for both GEMMs
#define M1         (B_N * T_N)    // 20736 = 324*64
#define M2         (B_N * NF_N)   // 17664 = 276*64

#define BM 64
#define BN 128
#define BK 32
#define TS 36                     // LDS tile stride (bank-conflict-free)

#define TWO_PI 6.283185307179586476925f

__device__ __forceinline__ float hann_f(int n) {
    return 0.5f - 0.5f * cosf((TWO_PI / WIN_N) * (float)n);
}

// Build both DFT matrices transposed, windows folded in. 2 x 1024 x 1024 f32.
//   WinvT[n][kk=2f+c] : irfft * (1/N) * synwin[n]   (rows n>=1022 are zero)
//   WfwdT[q][n]       : q<512 -> cos*hann, q>=512 -> -sin*hann (cols n>=1022 zero)
__global__ __launch_bounds__(256) void build_mats(float* __restrict__ WinvT,
                                                  float* __restrict__ WfwdT) {
    int idx = blockIdx.x * 256 + threadIdx.x;           // 0 .. 2*2^20-1
    int which = idx >> 20;
    int e   = idx & (KPAD * KPAD - 1);
    int row = e >> 10;
    int col = e & (KPAD - 1);
    if (which == 0) {
        float v = 0.f;
        int n = row;
        if (n < WIN_N) {
            int f = col >> 1, c = col & 1;
            int m = (f * n) % WIN_N;                    // exact angle reduction
            float ang = (TWO_PI / WIN_N) * (float)m;
            float wf  = (f == 0 || f == F_N - 1) ? 1.f : 2.f;
            float tr  = (c == 0) ? cosf(ang) : -sinf(ang);
            float h = hann_f(n);
            int p = n % HOP_N;
            float d = 0.f;
            for (int j = p; j < WIN_N; j += HOP_N) {
                float hj = hann_f(j);
                d += hj * hj;
            }
            v = wf * tr * (1.0f / (float)WIN_N) * (h / d);
        }
        WinvT[(size_t)row * KPAD + col] = v;
    } else {
        float v = 0.f;
        int n = col;
        if (n < WIN_N) {
            int k = row & (F_N - 1), c = row >> 9;
            int m = (k * n) % WIN_N;
            float ang = (TWO_PI / WIN_N) * (float)m;
            float tr  = (c == 0) ? cosf(ang) : -sinf(ang);
            v = tr * hann_f(n);
        }
        WfwdT[(size_t)row * KPAD + col] = v;
    }
}

// GEMM1: tf_frames = X(interleaved re/im) @ Winv, fused OLA via f32 atomics.
__global__ __launch_bounds__(256) void istft_gemm_ola(const float* __restrict__ x,
                                                      const float* __restrict__ WinvT,
                                                      float* __restrict__ sig) {
    __shared__ float sA[BM][TS];      // [m-row][k]
    __shared__ float sB[BN][TS];      // [n-col][k]  (W stored transposed)

    const int tid  = threadIdx.x;
    const int lane = tid & 31;
    const int wave = tid >> 5;
    const int wm   = wave & 1;          // 2 wave rows  (32 rows each)
    const int wn   = wave >> 1;         // 4 wave cols  (32 cols each)
    const int half = lane >> 4;
    const int r16  = lane & 15;

    const int rowBlock = blockIdx.y * BM;
    const int colBlock = blockIdx.x * BN;

    // A-tile loader: 4 float2 per thread; x[b,f,t,{re,im}] supplies K=2f,2f+1
    const int a_lr = tid & 63;
    const int a_lc = (tid >> 6) << 3;   // 0,8,16,24
    const int Ra   = rowBlock + a_lr;
    const int ab   = Ra / T_N;
    const int at   = Ra - ab * T_N;
    const float* xb = x + (size_t)ab * (F_N * T_N * 2) + at * 2;

    // B-tile loader: 4 float4 per thread from transposed W (coalesced)
    const int b_row = tid >> 3;         // 0..31 (+32,64,96)
    const int b_kc  = (tid & 7) << 2;   // 0,4,...,28

    v8f acc[2][2];
#pragma unroll
    for (int i = 0; i < 2; ++i)
#pragma unroll
        for (int j = 0; j < 2; ++j)
            acc[i][j] = (v8f){0.f,0.f,0.f,0.f,0.f,0.f,0.f,0.f};

    for (int kb = 0; kb < KPAD; kb += BK) {
        __syncthreads();
#pragma unroll
        for (int u = 0; u < 4; ++u) {
            int f = ((kb + a_lc) >> 1) + u;
            *(v2f*)&sA[a_lr][a_lc + (u << 1)] = *(const v2f*)(xb + (size_t)f * (T_N * 2));
        }
#pragma unroll
        for (int u = 0; u < 4; ++u) {
            int nrow = b_row + (u << 5);
            *(float4*)&sB[nrow][b_kc] =
                *(const float4*)(WinvT + (size_t)(colBlock + nrow) * KPAD + kb + b_kc);
        }
        __syncthreads();
#pragma unroll
        for (int k0 = 0; k0 < BK; k0 += 4) {
            const int ka = k0 + (half << 1);
            v2f a0 = *(const v2f*)&sA[(wm << 5) + r16][ka];
            v2f a1 = *(const v2f*)&sA[(wm << 5) + 16 + r16][ka];
            v2f b0 = *(const v2f*)&sB[(wn << 5) + r16][ka];
            v2f b1 = *(const v2f*)&sB[(wn << 5) + 16 + r16][ka];
            acc[0][0] = __builtin_amdgcn_wmma_f32_16x16x4_f32(false, a0, false, b0, (short)0, acc[0][0], false, false);
            acc[0][1] = __builtin_amdgcn_wmma_f32_16x16x4_f32(false, a0, false, b1, (short)0, acc[0][1], false, false);
            acc[1][0] = __builtin_amdgcn_wmma_f32_16x16x4_f32(false, a1, false, b0, (short)0, acc[1][0], false, false);
            acc[1][1] = __builtin_amdgcn_wmma_f32_16x16x4_f32(false, a1, false, b1, (short)0, acc[1][1], false, false);
        }
    }

    // Epilogue: overlap-add  sig[b, t*HOP + n] += tf
#pragma unroll
    for (int in = 0; in < 2; ++in) {
        const int n = colBlock + (wn << 5) + (in << 4) + r16;
        if (n < WIN_N) {
#pragma unroll
            for (int im = 0; im < 2; ++im) {
                const int Rb = rowBlock + (wm << 5) + (im << 4) + (half << 3);
#pragma unroll
                for (int v = 0; v < 8; ++v) {
                    int Rr = Rb + v;
                    int b  = Rr / T_N;
                    int t  = Rr - b * T_N;
                    atomicAdd(sig + (size_t)b * SIG_STRIDE + t * HOP_N + n, acc[im][in][v]);
                }
            }
        }
    }
}

// Segment crossfade fold (closed-form coefficients, no index tables).
__global__ __launch_bounds__(256) void fold_kernel(const float* __restrict__ sig,
                                                   float* __restrict__ outsig) {
    int i = blockIdx.x * 256 + threadIdx.x;
    if (i >= B_N * OUT_LEN) return;
    int b = i / OUT_LEN;
    int o = i - b * OUT_LEN;

    // invert FINAL_KEEP: output o -> kept index j = k*1000 + r
    int k = (o < 1000) ? 0 : (o - 50) / 950;
    int r = o - k * 950;
    int j = k * 1000 + r;

    const float* s = sig + (size_t)b * SIG_STRIDE;
    float kept = s[k * 1100 + r];                       // KEEP_IDX gather
    float sh = 0.f;
    int j2 = j + 50;
    if (j2 < KEPT_LEN) {
        int k2 = j2 / 1000, r2 = j2 - k2 * 1000;
        sh = s[k2 * 1100 + r2];
    }

    float cconst;
    if (j < 950)        cconst = 1.f;
    else if (j < 1000)  cconst = 0.f;
    else if (j < 29000) cconst = (r >= 50 && r < 950) ? 1.f : 0.f;
    else                cconst = 1.f;                   // j in [29050,30000)
    float cdec = (j < 29000 && r >= 950) ? (float)(1000 - r) * 0.02f : 0.f;
    float cinc = 0.f;
    if (j >= 950) {
        int rr = (j - 950) % 1000;
        if (rr < 50) cinc = (float)(rr + 1) * 0.02f;
    }
    outsig[(size_t)b * OUT_STRIDE + o] = (cconst + cdec) * kept + cinc * sh;
}

// GEMM2: S = frames(outsig) @ Wfwd, scattered into (B, 512, 69, 2) output.
__global__ __launch_bounds__(256) void stft_gemm(const float* __restrict__ outsig,
                                                 const float* __restrict__ WfwdT,
                                                 float* __restrict__ out) {
    __shared__ float sA[BM][TS];
    __shared__ float sB[BN][TS];

    const int tid  = threadIdx.x;
    const int lane = tid & 31;
    const int wave = tid >> 5;
    const int wm   = wave & 1;
    const int wn   = wave >> 1;
    const int half = lane >> 4;
    const int r16  = lane & 15;

    const int rowBlock = blockIdx.y * BM;
    const int colBlock = blockIdx.x * BN;

    // A-tile loader: two float4 per thread, coalesced along the frame
    const int a_lr = tid >> 2;          // 0..63
    const int a_lc = (tid & 3) << 3;    // 0,8,16,24
    const int Ra   = rowBlock + a_lr;
    const int ab   = Ra / NF_N;
    const int am   = Ra - ab * NF_N;
    const float* sg = outsig + (size_t)ab * OUT_STRIDE + am * HOP_N;

    const int b_row = tid >> 3;         // 0..31 (+32,64,96)
    const int b_kc  = (tid & 7) << 2;   // 0,4,...,28

    v8f acc[2][2];
#pragma unroll
    for (int i = 0; i < 2; ++i)
#pragma unroll
        for (int j = 0; j < 2; ++j)
            acc[i][j] = (v8f){0.f,0.f,0.f,0.f,0.f,0.f,0.f,0.f};

    for (int kb = 0; kb < KPAD; kb += BK) {
        __syncthreads();
        *(float4*)&sA[a_lr][a_lc]     = *(const float4*)(sg + kb + a_lc);
        *(float4*)&sA[a_lr][a_lc + 4] = *(const float4*)(sg + kb + a_lc + 4);
#pragma unroll
        for (int u = 0; u < 4; ++u) {
            int nrow = b_row + (u << 5);
            *(float4*)&sB[nrow][b_kc] =
                *(const float4*)(WfwdT + (size_t)(colBlock + nrow) * KPAD + kb + b_kc);
        }
        __syncthreads();
#pragma unroll
        for (int k0 = 0; k0 < BK; k0 += 4) {
            const int ka = k0 + (half << 1);
            v2f a0 = *(const v2f*)&sA[(wm << 5) + r16][ka];
            v2f a1 = *(const v2f*)&sA[(wm << 5) + 16 + r16][ka];
            v2f b0 = *(const v2f*)&sB[(wn << 5) + r16][ka];
            v2f b1 = *(const v2f*)&sB[(wn << 5) + 16 + r16][ka];
            acc[0][0] = __builtin_amdgcn_wmma_f32_16x16x4_f32(false, a0, false, b0, (short)0, acc[0][0], false, false);
            acc[0][1] = __builtin_amdgcn_wmma_f32_16x16x4_f32(false, a0, false, b1, (short)0, acc[0][1], false, false);
            acc[1][0] = __builtin_amdgcn_wmma_f32_16x16x4_f32(false, a1, false, b0, (short)0, acc[1][0], false, false);
            acc[1][1] = __builtin_amdgcn_wmma_f32_16x16x4_f32(false, a1, false, b1, (short)0, acc[1][1], false, false);
        }
    }

    // Epilogue: out[b, k, m, c] ; q = k + 512*c
#pragma unroll
    for (int in = 0; in < 2; ++in) {
        const int q  = colBlock + (wn << 5) + (in << 4) + r16;
        const int kf = q & (F_N - 1);
        const int cc = q >> 9;
#pragma unroll
        for (int im = 0; im < 2; ++im) {
            const int Rb = rowBlock + (wm << 5) + (im << 4) + (half << 3);
#pragma unroll
            for (int v = 0; v < 8; ++v) {
                int Rr = Rb + v;
                int b  = Rr / NF_N;
                int m  = Rr - b * NF_N;
                out[(size_t)b * (F_N * NF_N * 2) + kf * (NF_N * 2) + m * 2 + cc] = acc[im][in][v];
            }
        }
    }
}

extern "C" void kernel_launch(void* const* d_in, const int* in_sizes, int n_in,
                              void* d_out, int out_size, void* d_ws, size_t ws_size,
                              hipStream_t stream) {
    (void)in_sizes; (void)n_in; (void)out_size; (void)ws_size;
    const float* x = (const float*)d_in[0];
    float* out = (float*)d_out;

    float* ws     = (float*)d_ws;
    float* WinvT  = ws;                                   // 1M floats
    float* WfwdT  = WinvT + (size_t)KPAD * KPAD;          // 1M floats
    float* sig    = WfwdT + (size_t)KPAD * KPAD;          // B*SIG_STRIDE
    float* outsig = sig + (size_t)B_N * SIG_STRIDE;       // B*OUT_STRIDE

    hipMemsetAsync(sig, 0, (size_t)B_N * SIG_STRIDE * sizeof(float), stream);
    build_mats<<<(2 * KPAD * KPAD) / 256, 256, 0, stream>>>(WinvT, WfwdT);
    istft_gemm_ola<<<dim3(KPAD / BN, M1 / BM), 256, 0, stream>>>(x, WinvT, sig);
    fold_kernel<<<(B_N * OUT_LEN + 255) / 256, 256, 0, stream>>>(sig, outsig);
    stft_gemm<<<dim3(KPAD / BN, M2 / BM), 256, 0, stream>>>(outsig, WfwdT, out);
}